// GIN_87385404605077
// MI455X (gfx1250) — compile-verified
//
#include <hip/hip_runtime.h>
#include <hip/hip_bf16.h>
#include <math.h>

#define N_NODES 100000
#define N_EDGES 1600000
#define IN_DIM  128
#define DIM_H   64
#define N_GRAPHS 64
// 1/sqrt(1 + 1e-5)
#define BN_INVSTD 0.9999950000374997f

typedef __attribute__((ext_vector_type(2))) float v2f;
typedef __attribute__((ext_vector_type(8))) float v8f;

// ---------------------------------------------------------------- zero buffer
__global__ void gin_zero(float* __restrict__ p, unsigned n) {
  unsigned i = blockIdx.x * blockDim.x + threadIdx.x;
  if (i < n) p[i] = 0.0f;
}

// ------------------------------------------------- edge scatter-add (agg += h[src] at dst)
// one thread per (edge, feature); consecutive threads share an edge -> coalesced
__global__ void gin_scatter(const float* __restrict__ h,
                            const int* __restrict__ src,
                            const int* __restrict__ dst,
                            float* __restrict__ agg,
                            int dshift, unsigned total) {
  unsigned i = blockIdx.x * blockDim.x + threadIdx.x;
  if (i >= total) return;
  unsigned e = i >> dshift;
  unsigned f = i & ((1u << dshift) - 1u);
  int s = src[e];
  int d = dst[e];
  float v = h[((unsigned)s << dshift) + f];
  atomicAdd(&agg[((unsigned)d << dshift) + f], v);
}

// ------------------------------------------------- fused GIN MLP layer
// z = h + agg ; y = relu(((z@Wa + ba) * invstd) * g + be) ; out = relu(y@Wb + bb)
// block = 128 threads = 4 waves; each wave owns a 16-row x 64-col tile.
__global__ __launch_bounds__(128) void gin_layer(
    const float* __restrict__ h, const float* __restrict__ agg,
    const float* __restrict__ Wa, const float* __restrict__ ba,
    const float* __restrict__ gamma, const float* __restrict__ beta,
    const float* __restrict__ Wb, const float* __restrict__ bb,
    float* __restrict__ out, int K) {
  __shared__ float sWa[IN_DIM * DIM_H];   // up to 32KB (K x 64)
  __shared__ float sWb[DIM_H * DIM_H];    // 16KB
  __shared__ float sY[4 * 16 * DIM_H];    // 16KB : per-wave 16x64 staging

  const int tid = threadIdx.x;
  for (int i = tid; i < K * DIM_H; i += 128) sWa[i] = Wa[i];
  for (int i = tid; i < DIM_H * DIM_H; i += 128) sWb[i] = Wb[i];
  __syncthreads();

  const int wave = tid >> 5;
  const int lane = tid & 31;
  const int m    = lane & 15;   // M index within tile (and N index for B/C)
  const int kh   = lane >> 4;   // K half-select
  const int row0 = blockIdx.x * 64 + wave * 16;

  // clamp A-row so EXEC can stay all-ones for WMMA (stores are guarded)
  int rA = row0 + m;
  if (rA >= N_NODES) rA = N_NODES - 1;
  const float* __restrict__ hrow = h + (size_t)rA * K;
  const float* __restrict__ arow = agg + (size_t)rA * K;

  // ---------- GEMM 1: (h+agg) @ Wa, fp32 WMMA 16x16x4 ----------
  v8f c[4] = {};
  for (int k0 = 0; k0 < K; k0 += 4) {
    const int ka = k0 + 2 * kh;
    v2f hv = *(const v2f*)(hrow + ka);
    v2f av = *(const v2f*)(arow + ka);
    v2f a;
    a.x = hv.x + av.x;
    a.y = hv.y + av.y;
#pragma unroll
    for (int nt = 0; nt < 4; ++nt) {
      const int col = nt * 16 + m;
      v2f b;
      b.x = sWa[ka * DIM_H + col];
      b.y = sWa[(ka + 1) * DIM_H + col];
      c[nt] = __builtin_amdgcn_wmma_f32_16x16x4_f32(
          false, a, false, b, (short)0, c[nt], false, false);
    }
  }

  // ---------- epilogue 1: bias + BN(eval) + ReLU -> LDS staging ----------
  float* __restrict__ myY = sY + wave * (16 * DIM_H);
#pragma unroll
  for (int nt = 0; nt < 4; ++nt) {
    const int col = nt * 16 + m;
    const float bav = ba[col];
    const float gv  = gamma[col] * BN_INVSTD;
    const float bev = beta[col];
#pragma unroll
    for (int i = 0; i < 8; ++i) {
      float z = (c[nt][i] + bav) * gv + bev;
      z = fmaxf(z, 0.0f);
      myY[(i + 8 * kh) * DIM_H + col] = z;
    }
  }
  __syncthreads();

  // ---------- GEMM 2: y @ Wb ----------
  v8f d[4] = {};
  for (int k0 = 0; k0 < DIM_H; k0 += 4) {
    const int ka = k0 + 2 * kh;
    v2f a;
    a.x = myY[m * DIM_H + ka];
    a.y = myY[m * DIM_H + ka + 1];
#pragma unroll
    for (int nt = 0; nt < 4; ++nt) {
      const int col = nt * 16 + m;
      v2f b;
      b.x = sWb[ka * DIM_H + col];
      b.y = sWb[(ka + 1) * DIM_H + col];
      d[nt] = __builtin_amdgcn_wmma_f32_16x16x4_f32(
          false, a, false, b, (short)0, d[nt], false, false);
    }
  }

  // ---------- epilogue 2: bias + ReLU -> global ----------
#pragma unroll
  for (int nt = 0; nt < 4; ++nt) {
    const int col = nt * 16 + m;
    const float bbv = bb[col];
#pragma unroll
    for (int i = 0; i < 8; ++i) {
      const int row = row0 + i + 8 * kh;
      if (row < N_NODES) {
        out[(size_t)row * DIM_H + col] = fmaxf(d[nt][i] + bbv, 0.0f);
      }
    }
  }
}

// ------------------------------------------------- segment-sum pool (batch is sorted)
// thread (d, slot): walks a contiguous node segment for feature d, register-accumulates
// runs of equal graph id, flushes one atomic per run.
#define POOL_NPB 2048
#define POOL_SEG 512
__global__ __launch_bounds__(256) void gin_pool(const float* __restrict__ h,
                                                const int* __restrict__ batch,
                                                float* __restrict__ hp) {
  const int t = threadIdx.x;
  const int d = t & 63;
  const int slot = t >> 6;
  int start = blockIdx.x * POOL_NPB + slot * POOL_SEG;
  int end = start + POOL_SEG;
  if (end > N_NODES) end = N_NODES;
  int curg = -1;
  float acc = 0.0f;
  for (int i = start; i < end; ++i) {
    int g = batch[i];
    if (g != curg) {
      if (curg >= 0) atomicAdd(&hp[curg * DIM_H + d], acc);
      curg = g;
      acc = 0.0f;
    }
    acc += h[(size_t)i * DIM_H + d];
  }
  if (curg >= 0) atomicAdd(&hp[curg * DIM_H + d], acc);
}

// ------------------------------------------------- classifier head
// hc = [hp hp hp] ; z1 = relu(hc@Wl1+bl1) = relu(hp@(W1a+W1b+W1c)+bl1)
// z2 = z1@Wl2 + bl2 ; out = log_softmax(z2)
__global__ __launch_bounds__(64) void gin_head(const float* __restrict__ hp,
                                               const float* __restrict__ Wl1,
                                               const float* __restrict__ bl1,
                                               const float* __restrict__ Wl2,
                                               const float* __restrict__ bl2,
                                               float* __restrict__ outp) {
  const int r = threadIdx.x;  // graph id, 0..63
  float acc0 = bl2[0];
  float acc1 = bl2[1];
  for (int j = 0; j < 64; ++j) {
    float s = bl1[j];
    for (int k = 0; k < DIM_H; ++k) {
      float w = Wl1[k * 64 + j] + Wl1[(DIM_H + k) * 64 + j] + Wl1[(2 * DIM_H + k) * 64 + j];
      s += hp[r * DIM_H + k] * w;
    }
    s = fmaxf(s, 0.0f);
    acc0 += s * Wl2[j * 2 + 0];
    acc1 += s * Wl2[j * 2 + 1];
  }
  float mx = fmaxf(acc0, acc1);
  float lse = mx + logf(expf(acc0 - mx) + expf(acc1 - mx));
  outp[r * 2 + 0] = acc0 - lse;
  outp[r * 2 + 1] = acc1 - lse;
}

// ----------------------------------------------------------------------------
extern "C" void kernel_launch(void* const* d_in, const int* in_sizes, int n_in,
                              void* d_out, int out_size, void* d_ws, size_t ws_size,
                              hipStream_t stream) {
  const float* x     = (const float*)d_in[0];
  const int*   edge  = (const int*)d_in[1];      // [2, E] flat
  const int*   batch = (const int*)d_in[2];
  const float* W0a = (const float*)d_in[3];
  const float* b0a = (const float*)d_in[4];
  const float* g0  = (const float*)d_in[5];
  const float* be0 = (const float*)d_in[6];
  const float* W0b = (const float*)d_in[7];
  const float* b0b = (const float*)d_in[8];
  const float* Wsa = (const float*)d_in[9];
  const float* bsa = (const float*)d_in[10];
  const float* gs  = (const float*)d_in[11];
  const float* bes = (const float*)d_in[12];
  const float* Wsb = (const float*)d_in[13];
  const float* bsb = (const float*)d_in[14];
  const float* Wl1 = (const float*)d_in[15];
  const float* bl1 = (const float*)d_in[16];
  const float* Wl2 = (const float*)d_in[17];
  const float* bl2 = (const float*)d_in[18];
  float* outp = (float*)d_out;

  const int* src = edge;
  const int* dst = edge + N_EDGES;

  // workspace carve-up
  float* agg = (float*)d_ws;                        // N_NODES*IN_DIM
  float* hA  = agg + (size_t)N_NODES * IN_DIM;      // N_NODES*DIM_H
  float* hB  = hA + (size_t)N_NODES * DIM_H;        // N_NODES*DIM_H
  float* hp  = hB + (size_t)N_NODES * DIM_H;        // N_GRAPHS*DIM_H

  const int gemm_blocks = (N_NODES + 63) / 64;

  // ----- layer 0 (K = 128) -----
  {
    unsigned n = (unsigned)N_NODES * IN_DIM;
    gin_zero<<<(n + 255) / 256, 256, 0, stream>>>(agg, n);
    unsigned total = (unsigned)N_EDGES * IN_DIM;
    gin_scatter<<<(total + 255) / 256, 256, 0, stream>>>(x, src, dst, agg, 7, total);
    gin_layer<<<gemm_blocks, 128, 0, stream>>>(x, agg, W0a, b0a, g0, be0, W0b, b0b,
                                               hA, IN_DIM);
  }

  // ----- layers 1,2 (K = 64) -----
  const float* hin = hA;
  float* hout = hB;
  for (int l = 0; l < 2; ++l) {
    unsigned n = (unsigned)N_NODES * DIM_H;
    gin_zero<<<(n + 255) / 256, 256, 0, stream>>>(agg, n);
    unsigned total = (unsigned)N_EDGES * DIM_H;
    gin_scatter<<<(total + 255) / 256, 256, 0, stream>>>(hin, src, dst, agg, 6, total);
    gin_layer<<<gemm_blocks, 128, 0, stream>>>(
        hin, agg,
        Wsa + (size_t)l * DIM_H * DIM_H, bsa + l * DIM_H,
        gs + l * DIM_H, bes + l * DIM_H,
        Wsb + (size_t)l * DIM_H * DIM_H, bsb + l * DIM_H,
        hout, DIM_H);
    const float* t = hin; hin = hout; hout = (float*)t;
  }
  // after 2 swaps: hin points at final features (hA)

  // ----- pool + head -----
  {
    unsigned n = N_GRAPHS * DIM_H;
    gin_zero<<<1, 256, 0, stream>>>(hp, n);
    gin_pool<<<(N_NODES + POOL_NPB - 1) / POOL_NPB, 256, 0, stream>>>(hin, batch, hp);
    gin_head<<<1, 64, 0, stream>>>(hp, Wl1, bl1, Wl2, bl2, outp);
  }
}